// AdaptiveSparseAttention_24859270709416
// MI455X (gfx1250) — compile-verified
//
#include <hip/hip_runtime.h>
#include <hip/hip_bf16.h>

typedef float v2f __attribute__((ext_vector_type(2)));
typedef float v8f __attribute__((ext_vector_type(8)));

#define NB 4
#define NL 1024
#define NDIM 512
#define NHEADS 8
#define NHD 64
#define NHID 256
#define KK 716          // max(1, int(1024*0.7))
#define SCST 1025       // padded LDS row stride (odd -> no bank conflicts)

__device__ __forceinline__ v8f wmma4(v2f a, v2f b, v8f c) {
    return __builtin_amdgcn_wmma_f32_16x16x4_f32(false, a, false, b, (short)0, c, false, false);
}

__device__ __forceinline__ v8f v8zero() {
    v8f z;
#pragma unroll
    for (int i = 0; i < 8; ++i) z[i] = 0.0f;
    return z;
}

__device__ __forceinline__ unsigned f2ord(float f) {
    unsigned u = __float_as_uint(f);
    return (u & 0x80000000u) ? ~u : (u | 0x80000000u);
}

__device__ __forceinline__ int wred_sum_i(int v) {
#pragma unroll
    for (int o = 16; o > 0; o >>= 1) v += __shfl_xor(v, o, 32);
    return v;
}
__device__ __forceinline__ float wred_sum_f(float v) {
#pragma unroll
    for (int o = 16; o > 0; o >>= 1) v += __shfl_xor(v, o, 32);
    return v;
}
__device__ __forceinline__ float wred_max_f(float v) {
#pragma unroll
    for (int o = 16; o > 0; o >>= 1) v = fmaxf(v, __shfl_xor(v, o, 32));
    return v;
}

// ---------------- Kernel 1: pattern selector MLP (per batch) ----------------
__global__ __launch_bounds__(256) void sel_kernel(const float* __restrict__ x,
                                                  const float* __restrict__ W1,
                                                  const float* __restrict__ b1,
                                                  const float* __restrict__ W2,
                                                  const float* __restrict__ b2,
                                                  const float* __restrict__ lpt,
                                                  float* __restrict__ pwout) {
    __shared__ float pooled[NDIM];
    __shared__ float hid[NHID];
    __shared__ float lg[3];
    int b = blockIdx.x, t = threadIdx.x;
    for (int d = t; d < NDIM; d += 256) {
        float s = 0.0f;
        const float* xp = x + (size_t)b * NL * NDIM + d;
        for (int i = 0; i < NL; ++i) s += xp[(size_t)i * NDIM];
        pooled[d] = s * (1.0f / (float)NL);
    }
    __syncthreads();
    {
        float s = b1[t];
        const float* wp = W1 + (size_t)t * NDIM;
        for (int k = 0; k < NDIM; ++k) s += wp[k] * pooled[k];
        hid[t] = fmaxf(s, 0.0f);
    }
    __syncthreads();
    if (t < 3) {
        float s = b2[t];
        const float* wp = W2 + (size_t)t * NHID;
        for (int j = 0; j < NHID; ++j) s += wp[j] * hid[j];
        lg[t] = s;
    }
    __syncthreads();
    if (t == 0) {
        float tau = fminf(fmaxf(expf(lpt[0]), 1e-4f), 10.0f);
        float l0 = lg[0] / tau, l1 = lg[1] / tau, l2 = lg[2] / tau;
        float m = fmaxf(l0, fmaxf(l1, l2));
        float e0 = expf(l0 - m), e1 = expf(l1 - m), e2 = expf(l2 - m);
        float inv = 1.0f / (e0 + e1 + e2);
        pwout[b * 4 + 0] = e0 * inv;
        pwout[b * 4 + 1] = e1 * inv;
        pwout[b * 4 + 2] = e2 * inv;
    }
}

// ---------------- Kernel 2: QKV GEMM (x @ Wqkv^T), f32 WMMA ----------------
// One wave per 16(M) x 64(N) tile. q/k/v written as (B,H,L,hd).
__global__ __launch_bounds__(32) void qkv_kernel(const float* __restrict__ x,
                                                 const float* __restrict__ w,
                                                 float* __restrict__ q,
                                                 float* __restrict__ k,
                                                 float* __restrict__ v) {
    int lane = threadIdx.x, lr = lane & 15, half = lane >> 4;
    int mt = blockIdx.x / 24, nb = blockIdx.x % 24;
    int m0 = mt * 16, n0 = nb * 64;
    v8f acc[4];
#pragma unroll
    for (int t = 0; t < 4; ++t) acc[t] = v8zero();
    const float* arow = x + (size_t)(m0 + lr) * NDIM;
    for (int k0 = 0; k0 < NDIM; k0 += 4) {
        int kidx = k0 + 2 * half;
        v2f a; a[0] = arow[kidx]; a[1] = arow[kidx + 1];
#pragma unroll
        for (int t = 0; t < 4; ++t) {
            const float* brow = w + (size_t)(n0 + 16 * t + lr) * NDIM + kidx;
            v2f bb; bb[0] = brow[0]; bb[1] = brow[1];
            acc[t] = wmma4(a, bb, acc[t]);
        }
    }
    int bidx = m0 >> 10, il = m0 & (NL - 1);
#pragma unroll
    for (int t = 0; t < 4; ++t) {
        int n = n0 + 16 * t + lr;
        int sel = n >> 9, hn = n & 511, h = hn >> 6, d = hn & 63;
        float* base = (sel == 0) ? q : ((sel == 1) ? k : v);
        float* dst = base + ((size_t)(bidx * NHEADS + h) * NL) * NHD + d;
#pragma unroll
        for (int r = 0; r < 8; ++r) {
            int i = il + r + 8 * half;
            dst[(size_t)i * NHD] = acc[t][r];
        }
    }
}

// ---- Kernel 3: fused scores + exact top-k gate + softmax + attn@V ----
__global__ __launch_bounds__(256) void attn_kernel(const float* __restrict__ qw,
                                                   const float* __restrict__ kw,
                                                   const float* __restrict__ vw,
                                                   const float* __restrict__ pw,
                                                   const int* __restrict__ mask,
                                                   const float* __restrict__ spw,
                                                   const float* __restrict__ spb,
                                                   float* __restrict__ ctx) {
    __shared__ float sc[16 * SCST];
    int tid = threadIdx.x, w = tid >> 5, lane = tid & 31, lr = lane & 15, half = lane >> 4;
    int bh = blockIdx.x >> 6, it = blockIdx.x & 63, i0 = it * 16;
    int b = bh >> 3, h = bh & 7;
    const float* qb = qw + (size_t)bh * NL * NHD;
    const float* kb = kw + (size_t)bh * NL * NHD;
    const float* vb = vw + (size_t)bh * NL * NHD;
    const float scale = 0.125f;  // 64^-0.5

    // Phase 1: 16 x 1024 score strip; wave w covers columns [w*128, w*128+128)
    {
        v8f acc[8];
#pragma unroll
        for (int t = 0; t < 8; ++t) acc[t] = v8zero();
        for (int k0 = 0; k0 < NHD; k0 += 4) {
            int kidx = k0 + 2 * half;
            const float* ap = qb + (size_t)(i0 + lr) * NHD + kidx;
            v2f a; a[0] = ap[0]; a[1] = ap[1];
#pragma unroll
            for (int t = 0; t < 8; ++t) {
                int j = w * 128 + t * 16 + lr;
                const float* bp = kb + (size_t)j * NHD + kidx;
                v2f bb; bb[0] = bp[0]; bb[1] = bp[1];
                acc[t] = wmma4(a, bb, acc[t]);
            }
        }
#pragma unroll
        for (int t = 0; t < 8; ++t) {
            int j = w * 128 + t * 16 + lr;
#pragma unroll
            for (int r = 0; r < 8; ++r) sc[(r + 8 * half) * SCST + j] = acc[t][r] * scale;
        }
    }
    __syncthreads();

    // Phase 2: each wave fully owns rows 2w and 2w+1 (1024 elems = 32/lane)
    float p0 = pw[b * 4 + 0], p1 = pw[b * 4 + 1], p2 = pw[b * 4 + 2];
    float sw_ = spw[h], sb_ = spb[h];
    const int* mrow = mask + (size_t)b * NL;
    for (int rr = 0; rr < 2; ++rr) {
        int ri = 2 * w + rr;
        int i = i0 + ri;
        float fv[32];
        unsigned uv[32];
#pragma unroll
        for (int t = 0; t < 32; ++t) {
            fv[t] = sc[ri * SCST + lane + 32 * t];
            uv[t] = f2ord(fv[t] * sw_ + sb_);
        }
        // exact kth-largest of s via bitwise binary search on sortable uints
        unsigned K = 0u;
        for (int bit = 31; bit >= 0; --bit) {
            unsigned cand = K | (1u << bit);
            int c = 0;
#pragma unroll
            for (int t = 0; t < 32; ++t) c += (uv[t] >= cand) ? 1 : 0;
            c = wred_sum_i(c);
            if (c >= KK) K = cand;
        }
        // gate: combined = pw0*local + pw1 + pw2*smask > 0.05  && mask
        float mx = -3.4e38f;
#pragma unroll
        for (int t = 0; t < 32; ++t) {
            int j = lane + 32 * t;
            float comb = p1;
            if (j >= i - 16 && j <= i + 16) comb += p0;
            if (uv[t] >= K) comb += p2;
            bool allowed = (comb > 0.05f) && (mrow[j] != 0);
            float val = allowed ? fv[t] : -1e9f;
            fv[t] = val;
            mx = fmaxf(mx, val);
        }
        mx = wred_max_f(mx);
        bool allm = (mx <= -1e9f + 1e-6f);
        if (allm) { mx = 0.0f; if (lane == 0) fv[0] = 0.0f; }
        float s = 0.0f;
#pragma unroll
        for (int t = 0; t < 32; ++t) { float e = expf(fv[t] - mx); fv[t] = e; s += e; }
        s = wred_sum_f(s);
        float inv = 1.0f / s;
#pragma unroll
        for (int t = 0; t < 32; ++t) sc[ri * SCST + lane + 32 * t] = fv[t] * inv;
    }
    __syncthreads();

    // Phase 3: attn (16x1024, LDS) @ V (1024x64). Waves 0..3 each do one
    // 16x16 column tile over full K -> deterministic (no float atomics).
    if (w < 4) {
        v8f acc = v8zero();
        int d = w * 16 + lr;
        for (int k0 = 0; k0 < NL; k0 += 4) {
            int kidx = k0 + 2 * half;
            v2f a; a[0] = sc[lr * SCST + kidx]; a[1] = sc[lr * SCST + kidx + 1];
            v2f bb; bb[0] = vb[(size_t)kidx * NHD + d]; bb[1] = vb[(size_t)(kidx + 1) * NHD + d];
            acc = wmma4(a, bb, acc);
        }
        float* cbase = ctx + (size_t)bh * NL * NHD;
#pragma unroll
        for (int r = 0; r < 8; ++r)
            cbase[(size_t)(i0 + r + 8 * half) * NHD + d] = acc[r];
    }
}

// ---------------- Kernel 4: output projection (ctx @ Wproj^T + b) ----------------
__global__ __launch_bounds__(32) void proj_kernel(const float* __restrict__ ctx,
                                                  const float* __restrict__ w,
                                                  const float* __restrict__ bias,
                                                  float* __restrict__ out) {
    int lane = threadIdx.x, lr = lane & 15, half = lane >> 4;
    int mt = blockIdx.x >> 3, nb = blockIdx.x & 7;
    int m0 = mt * 16, n0 = nb * 64;
    int b = m0 >> 10, il = m0 & (NL - 1);
    v8f acc[4];
#pragma unroll
    for (int t = 0; t < 4; ++t) acc[t] = v8zero();
    for (int k0 = 0; k0 < NDIM; k0 += 4) {
        int kidx = k0 + 2 * half;
        int hh = kidx >> 6, dd = kidx & 63;  // kidx even -> kidx+1 stays in same head
        const float* ap = ctx + ((size_t)(b * NHEADS + hh) * NL + il + lr) * NHD + dd;
        v2f a; a[0] = ap[0]; a[1] = ap[1];
#pragma unroll
        for (int t = 0; t < 4; ++t) {
            const float* bp = w + (size_t)(n0 + 16 * t + lr) * NDIM + kidx;
            v2f bb; bb[0] = bp[0]; bb[1] = bp[1];
            acc[t] = wmma4(a, bb, acc[t]);
        }
    }
#pragma unroll
    for (int t = 0; t < 4; ++t) {
        int n = n0 + 16 * t + lr;
        float bv = bias[n];
#pragma unroll
        for (int r = 0; r < 8; ++r) {
            int m = m0 + r + 8 * half;
            out[(size_t)m * NDIM + n] = acc[t][r] + bv;
        }
    }
}

extern "C" void kernel_launch(void* const* d_in, const int* in_sizes, int n_in,
                              void* d_out, int out_size, void* d_ws, size_t ws_size,
                              hipStream_t stream) {
    const float* x     = (const float*)d_in[0];
    const int*   mask  = (const int*)d_in[1];
    const float* Wqkv  = (const float*)d_in[2];
    const float* Wproj = (const float*)d_in[3];
    const float* bproj = (const float*)d_in[4];
    const float* Wsel1 = (const float*)d_in[5];
    const float* bsel1 = (const float*)d_in[6];
    const float* Wsel2 = (const float*)d_in[7];
    const float* bsel2 = (const float*)d_in[8];
    const float* lpt   = (const float*)d_in[9];
    const float* spw   = (const float*)d_in[10];
    const float* spb   = (const float*)d_in[11];
    float* out = (float*)d_out;

    const size_t perT = (size_t)NB * NHEADS * NL * NHD;  // 2M floats
    float* ws  = (float*)d_ws;
    float* pw  = ws;            // 16 floats
    float* q   = ws + 16;
    float* k   = q + perT;
    float* v   = k + perT;
    float* ctx = v + perT;

    sel_kernel<<<NB, 256, 0, stream>>>(x, Wsel1, bsel1, Wsel2, bsel2, lpt, pw);
    qkv_kernel<<<(NB * NL / 16) * 24, 32, 0, stream>>>(x, Wqkv, q, k, v);
    attn_kernel<<<NB * NHEADS * (NL / 16), 256, 0, stream>>>(q, k, v, pw, mask, spw, spb, ctx);
    proj_kernel<<<(NB * NL / 16) * (NDIM / 64), 32, 0, stream>>>(ctx, Wproj, bproj, out);
}